// MagicNumberInterpolation_55009941127452
// MI455X (gfx1250) — compile-verified
//
#include <hip/hip_runtime.h>
#include <stdint.h>

// ---------------------------------------------------------------------------
// MagicNumberInterpolation for MI455X (gfx1250, wave32).
//
// Roofline: 64 MiB in + 64 MiB out = 128 MiB minimum HBM traffic -> ~5.5 us
// @ 23.3 TB/s. Memory-bound run-length scan; no matrix math -> no WMMA.
// CDNA5 features: ALL bulk HBM reads go through the Tensor Data Mover
// (tensor_load_to_lds, TENSORcnt), bulk writes go through async LDS->global
// b128 stores (ASYNCcnt); global_prefetch warms GL2; the 192 MB L2 absorbs
// the 16 MB of chunk summaries and the K3 re-read of x.
//
// 3-phase segmented scan (T split into 128 chunks of 32 for parallelism:
// 524288 threads, vs only 4096 natural columns):
//   K1: per block: TDM-stage a 32x256 f32 tile to LDS; per-thread first/last
//       non-magic summary of its column-chunk.
//   K2: per column: prefix (prev) + suffix (next) chain over chunk summaries.
//   K3: per block: TDM-stage the tile again (L2 hit), forward prev-sweep +
//       backward next-sweep fused with the lerp, result written back into the
//       LDS tile, then async-stored to global in b128 segments.
// ---------------------------------------------------------------------------

#define MAGIC 0.0f

constexpr unsigned Bc = 8;
constexpr unsigned Tc = 4096;
constexpr unsigned Dc = 512;          // 2^9
constexpr unsigned Cc = Bc * Dc;      // 4096 columns, 2^12
constexpr unsigned CT = 32;           // chunk length along T
constexpr unsigned NC = Tc / CT;      // 128 chunks per column
constexpr unsigned NCC = NC * Cc;     // 524288 (column,chunk) pairs
constexpr unsigned BLOCK = 256;

typedef uint32_t u32x4 __attribute__((ext_vector_type(4)));
typedef uint32_t u32x8 __attribute__((ext_vector_type(8)));

// ---------------------------------------------------------------------------
// Tensor Data Mover: DMA one CT x BLOCK f32 tile (rows 1 KiB contiguous,
// row stride Dc*4 = 2 KiB) from global into LDS. Call from ONE wave only
// (TDM issues per-wave regardless of EXEC); caller barriers afterwards.
// D# group0: count=1 | lds_addr | global_addr[56:0] | type=2.
// D# group1: data_size=4B, tensor_dim0=Dc, tensor_dim1=Bc*Tc,
//            tile_dim0=BLOCK, tile_dim1=CT, tensor_dim0_stride=Dc.
// ---------------------------------------------------------------------------
__device__ __forceinline__ void tdm_load_tile(uint32_t lds_addr, uint64_t ga) {
  u32x4 g0;
  g0.x = 1u;                                            // count=1, user D#
  g0.y = lds_addr;                                      // LDS byte address
  g0.z = (uint32_t)ga;                                  // global_addr[31:0]
  g0.w = ((uint32_t)(ga >> 32) & 0x01FFFFFFu) | 0x80000000u; // [56:32]|type=2
  u32x8 g1;
  g1.s0 = 2u << 16;                                     // wg_mask=0, 4B elems
  g1.s1 = (Dc & 0xFFFFu) << 16;                         // tensor_dim0 lo16
  g1.s2 = (Dc >> 16) | (((Tc * Bc) & 0xFFFFu) << 16);   // dim0 hi | dim1 lo
  g1.s3 = ((Tc * Bc) >> 16) | (BLOCK << 16);            // dim1 hi | tile_dim0
  g1.s4 = CT;                                           // tile_dim1, tile_dim2=0
  g1.s5 = Dc;                                           // tensor_dim0_stride
  g1.s6 = 0u;
  g1.s7 = 0u;
  asm volatile("tensor_load_to_lds %0, %1" :: "s"(g0), "s"(g1) : "memory");
  __builtin_amdgcn_s_wait_tensorcnt(0);
}

// ---------------------------------------------------------------------------
// K1: TDM tile -> LDS, then per-thread first/last non-magic chunk summary.
// ---------------------------------------------------------------------------
__global__ void k_summary(const float* __restrict__ x,
                          float* __restrict__ last_val, int* __restrict__ last_idx,
                          float* __restrict__ first_val, int* __restrict__ first_idx) {
  extern __shared__ unsigned char smem[];
  float* s_x = (float*)smem;                            // [CT*BLOCK]

  const uint32_t tid = threadIdx.x;
  const uint32_t g = blockIdx.x * BLOCK + tid;          // < NCC
  const uint32_t k = g >> 12;                           // chunk (block-uniform)
  const uint32_t t0 = k * CT;

  const uint32_t c0 = (blockIdx.x * BLOCK) & (Cc - 1u); // block's first column
  const uint32_t b0 = c0 >> 9;
  const uint32_t d0 = c0 & (Dc - 1u);
  const size_t tile_off = (size_t)b0 * Tc * Dc + (size_t)t0 * Dc + d0;

  // Speculative GL2 warm-up for the next chunk's tile (global_prefetch_b8).
  __builtin_prefetch(x + tile_off + (size_t)CT * Dc, 0, 1);

  if (tid < 32) {
    tdm_load_tile((uint32_t)(uintptr_t)s_x, (uint64_t)(uintptr_t)(x + tile_off));
  }
  __syncthreads();

  int lp = -1, fp = (int)Tc;
  float lv = MAGIC, fv = MAGIC;
#pragma unroll
  for (uint32_t i = 0; i < CT; ++i) {
    const float v = s_x[i * BLOCK + tid];
    if (v != MAGIC) {
      if (fp == (int)Tc) { fp = (int)(t0 + i); fv = v; }
      lp = (int)(t0 + i);
      lv = v;
    }
  }
  last_val[g] = lv;
  last_idx[g] = lp;
  first_val[g] = fv;
  first_idx[g] = fp;
}

// ---------------------------------------------------------------------------
// K2: per-column exclusive prefix (prev chain) + suffix (next chain) over the
// chunk summaries. ~16 MB traffic, L2-resident.
// ---------------------------------------------------------------------------
__global__ void k_chain(const float* __restrict__ last_val, const int* __restrict__ last_idx,
                        const float* __restrict__ first_val, const int* __restrict__ first_idx,
                        float* __restrict__ pre_val, int* __restrict__ pre_idx,
                        float* __restrict__ suf_val, int* __restrict__ suf_idx) {
  const uint32_t c = blockIdx.x * BLOCK + threadIdx.x;  // < Cc
  int p = -1;
  float v = MAGIC;
  for (uint32_t k = 0; k < NC; ++k) {
    const uint32_t g = k * Cc + c;
    pre_idx[g] = p;
    pre_val[g] = v;
    const int li = last_idx[g];
    if (li >= 0) { p = li; v = last_val[g]; }
  }
  int n = (int)Tc;
  float ev = MAGIC;
  for (int k = (int)NC - 1; k >= 0; --k) {
    const uint32_t g = (uint32_t)k * Cc + c;
    suf_idx[g] = n;
    suf_val[g] = ev;
    const int fi = first_idx[g];
    if (fi < (int)Tc) { n = fi; ev = first_val[g]; }
  }
}

// ---------------------------------------------------------------------------
// K3: TDM tile load -> LDS sweeps -> async b128 tile store.
// Dynamic LDS: s_x[CT][BLOCK] f32 (32KB, doubles as out tile) +
// s_pv[CT][BLOCK] f32 (32KB) + s_pi[CT][BLOCK] i16 (16KB) = 80KB
// -> 4 workgroups per 320KB WGP.
// ---------------------------------------------------------------------------
__global__ void k_final(const float* __restrict__ x,
                        const float* __restrict__ pre_val, const int* __restrict__ pre_idx,
                        const float* __restrict__ suf_val, const int* __restrict__ suf_idx,
                        float* __restrict__ out) {
  extern __shared__ unsigned char smem[];
  float* s_x = (float*)smem;                            // [CT*BLOCK] in->out
  float* s_pv = s_x + CT * BLOCK;                       // [CT*BLOCK]
  short* s_pi = (short*)(s_pv + CT * BLOCK);            // [CT*BLOCK]

  const uint32_t tid = threadIdx.x;
  const uint32_t g = blockIdx.x * BLOCK + tid;          // < NCC
  const uint32_t k = g >> 12;                           // chunk (block-uniform)
  const uint32_t t0 = k * CT;

  const uint32_t c0 = (blockIdx.x * BLOCK) & (Cc - 1u);
  const uint32_t b0 = c0 >> 9;
  const uint32_t d0 = c0 & (Dc - 1u);
  const size_t tile_off = (size_t)b0 * Tc * Dc + (size_t)t0 * Dc + d0;
  const uint32_t lds_x0 = (uint32_t)(uintptr_t)s_x;     // LDS byte offset

  if (tid < 32) {
    tdm_load_tile(lds_x0, (uint64_t)(uintptr_t)(x + tile_off));
  }
  __syncthreads();                                      // tile visible to all

  // --- Forward sweep: running last-non-magic (p, sv) per column. ---
  int p = pre_idx[g];
  float sv = pre_val[g];
#pragma unroll
  for (uint32_t i = 0; i < CT; ++i) {
    const float v = s_x[i * BLOCK + tid];
    if (v != MAGIC) { p = (int)(t0 + i); sv = v; }
    s_pv[i * BLOCK + tid] = sv;
    s_pi[i * BLOCK + tid] = (short)p;                   // p in [-1, 4095]
  }

  // --- Backward sweep: running next-non-magic (n, ev); fused lerp; result
  // overwrites the tile slot consumed this iteration. ---
  int n = suf_idx[g];
  float ev = suf_val[g];
#pragma unroll
  for (int i = (int)CT - 1; i >= 0; --i) {
    const int t = (int)t0 + i;
    const float v = s_x[(uint32_t)i * BLOCK + tid];
    float o;
    if (v != MAGIC) {
      n = t;
      ev = v;
      o = v;
    } else {
      const float svt = s_pv[(uint32_t)i * BLOCK + tid];
      const int pt = (int)s_pi[(uint32_t)i * BLOCK + tid];
      const bool hp = pt >= 0;
      const bool hn = n < (int)Tc;
      const float start = hp ? svt : ev;                // leading -> next val
      const float end = hn ? ev : svt;                  // trailing -> prev val
      const float w = (hp && hn) ? (float)(t - pt) / (float)(n - pt) : 0.0f;
      o = start + w * (end - start);
    }
    s_x[(uint32_t)i * BLOCK + tid] = o;                 // tile holds output
  }
  __syncthreads();                                      // all columns done

  // --- Async store LDS tile -> out, 16B/lane b128 segments, coalesced. ---
  const uint64_t ga_out = (uint64_t)(uintptr_t)(out + tile_off);
  constexpr uint32_t SEGS_PER_ROW = BLOCK * 4 / 16;     // 64
  constexpr uint32_t SEGS = CT * SEGS_PER_ROW;          // 2048
#pragma unroll
  for (uint32_t s = 0; s < SEGS / BLOCK; ++s) {         // 8 per thread
    const uint32_t seg = s * BLOCK + tid;
    const uint32_t row = seg >> 6;                      // / SEGS_PER_ROW
    const uint32_t o16 = seg & (SEGS_PER_ROW - 1u);
    const uint64_t ga = ga_out + (uint64_t)row * (Dc * 4) + (uint64_t)o16 * 16;
    const uint32_t la = lds_x0 + row * (BLOCK * 4) + o16 * 16;
    asm volatile("global_store_async_from_lds_b128 %0, %1, off"
                 :: "v"(ga), "v"(la)
                 : "memory");
  }
  asm volatile("s_wait_asynccnt 0" ::: "memory");       // (ENDPGM waits too)
}

// ---------------------------------------------------------------------------
// Fallback (no workspace): one thread per column, run-length fill.
// ---------------------------------------------------------------------------
__global__ void k_fallback(const float* __restrict__ x, float* __restrict__ out) {
  const uint32_t c = blockIdx.x * BLOCK + threadIdx.x;
  if (c >= Cc) return;
  const uint32_t b = c >> 9, d = c & (Dc - 1u);
  const float* xc = x + (size_t)b * Tc * Dc + d;
  float* oc = out + (size_t)b * Tc * Dc + d;

  int p = -1;
  float sv = MAGIC;
  int t = 0;
  while (t < (int)Tc) {
    const float v = xc[(size_t)t * Dc];
    if (v != MAGIC) {
      oc[(size_t)t * Dc] = v;
      p = t;
      sv = v;
      ++t;
      continue;
    }
    int n = t + 1;
    float ev = MAGIC;
    while (n < (int)Tc) {
      const float u = xc[(size_t)n * Dc];
      if (u != MAGIC) { ev = u; break; }
      ++n;
    }
    const bool hp = p >= 0;
    const bool hn = n < (int)Tc;
    const float start = hp ? sv : ev;
    const float end = hn ? ev : sv;
    for (int j = t; j < n; ++j) {
      const float w = (hp && hn) ? (float)(j - p) / (float)(n - p) : 0.0f;
      oc[(size_t)j * Dc] = start + w * (end - start);
    }
    t = n;
  }
}

// ---------------------------------------------------------------------------
extern "C" void kernel_launch(void* const* d_in, const int* in_sizes, int n_in,
                              void* d_out, int out_size, void* d_ws, size_t ws_size,
                              hipStream_t stream) {
  const float* x = (const float*)d_in[0];
  float* out = (float*)d_out;

  const size_t need = (size_t)NCC * 4u * 8u;            // 16 MB of summaries
  if (d_ws != nullptr && ws_size >= need) {
    float* last_val = (float*)d_ws;
    float* first_val = last_val + NCC;
    float* pre_val = first_val + NCC;
    float* suf_val = pre_val + NCC;
    int* last_idx = (int*)(suf_val + NCC);
    int* first_idx = last_idx + NCC;
    int* pre_idx = first_idx + NCC;
    int* suf_idx = pre_idx + NCC;

    const size_t smem1 = (size_t)CT * BLOCK * 4;        // 32 KB (tile only)
    k_summary<<<NCC / BLOCK, BLOCK, smem1, stream>>>(x, last_val, last_idx,
                                                     first_val, first_idx);
    k_chain<<<Cc / BLOCK, BLOCK, 0, stream>>>(last_val, last_idx, first_val, first_idx,
                                              pre_val, pre_idx, suf_val, suf_idx);
    const size_t smem3 = (size_t)CT * BLOCK * (4 + 4 + 2);  // 80 KB
    k_final<<<NCC / BLOCK, BLOCK, smem3, stream>>>(x, pre_val, pre_idx,
                                                   suf_val, suf_idx, out);
  } else {
    k_fallback<<<(Cc + BLOCK - 1) / BLOCK, BLOCK, 0, stream>>>(x, out);
  }
}